// MultiHeadAttention_39676907883716
// MI455X (gfx1250) — compile-verified
//
#include <hip/hip_runtime.h>

constexpr int DMODEL = 6144;
constexpr int NHEAD  = 2048;
constexpr int KS     = 4;              // K-split factor for extra wave parallelism
constexpr int KLEN   = DMODEL / KS;    // 1536 K per wave

typedef float v2f __attribute__((ext_vector_type(2)));
typedef float v4f __attribute__((ext_vector_type(4)));
typedef float v8f __attribute__((ext_vector_type(8)));

// ---------------------------------------------------------------------------
// Stage A: partial qkv:  qkvp[(mat*KS + kz)*D + row] = sum_{k in slice} W[row][k]*x[k]
// One wave32 computes 16 rows over one K slice via V_WMMA_F32_16X16X4_F32.
// A tile: lanes 0-15 hold W[row][k..k+3], lanes 16-31 hold W[row][k+4..k+7]
// (one NT 128-bit load per lane; every weight byte touched exactly once).
// B = x replicated over all 16 columns -> every tile column holds the GEMV.
// ---------------------------------------------------------------------------
__global__ __launch_bounds__(128) void gemv_qkv_wmma(
    const float* __restrict__ x,
    const float* __restrict__ Wq,
    const float* __restrict__ Wk,
    const float* __restrict__ Wv,
    float* __restrict__ qkvp)
{
    const int wave = threadIdx.x >> 5;
    const int lane = threadIdx.x & 31;
    const int half = lane >> 4;          // 0 -> K slots 0,1 ; 1 -> K slots 2,3
    const int mrow = lane & 15;
    const int m0   = (blockIdx.x * 4 + wave) * 16;
    const int mat  = blockIdx.y;         // 0=q, 1=k, 2=v
    const int kz   = blockIdx.z;         // K slice
    const float* W = (mat == 0) ? Wq : (mat == 1) ? Wk : Wv;

    const float* wrow = W + (size_t)(m0 + mrow) * DMODEL + half * 4;
    const int kbeg = kz * KLEN;

    v8f c = {};
    #pragma unroll 8
    for (int k0 = kbeg; k0 < kbeg + KLEN; k0 += 8) {
        v4f a  = __builtin_nontemporal_load((const v4f*)(wrow + k0)); // streaming
        v4f xl = *(const v4f*)(x + k0);       // uniform -> scalar loads
        v4f xh = *(const v4f*)(x + k0 + 4);
        v2f a0; a0[0] = a[0]; a0[1] = a[1];
        v2f a1; a1[0] = a[2]; a1[1] = a[3];
        v2f b0; b0[0] = half ? xh[0] : xl[0]; b0[1] = half ? xh[1] : xl[1];
        v2f b1; b1[0] = half ? xh[2] : xl[2]; b1[1] = half ? xh[3] : xl[3];
        c = __builtin_amdgcn_wmma_f32_16x16x4_f32(false, a0, false, b0,
                                                  (short)0, c, false, false);
        c = __builtin_amdgcn_wmma_f32_16x16x4_f32(false, a1, false, b1,
                                                  (short)0, c, false, false);
    }
    // All 16 tile columns identical; lane 0 holds rows m0..m0+7 (VGPR v -> M=v),
    // lane 16 holds rows m0+8..m0+15.
    if ((lane & 15) == 0) {
        float* out = qkvp + (size_t)(mat * KS + kz) * DMODEL + m0 + half * 8;
        #pragma unroll
        for (int v = 0; v < 8; ++v) out[v] = c[v];
    }
}

// ---------------------------------------------------------------------------
// Stage B: reduce the K-split partials, then per-head 3x3:
// M = [q_h; k_h; v_h], G = (M M^T)/sqrt(3), O = G M.
// y is the flat (3*6144) view: y[h*9 + i*3 + l] = O[i][l].
// ---------------------------------------------------------------------------
__global__ __launch_bounds__(256) void head_attn_3x3(
    const float* __restrict__ qkvp, float* __restrict__ y)
{
    const int h = blockIdx.x * blockDim.x + threadIdx.x;
    if (h >= NHEAD) return;
    float M[3][3];
    #pragma unroll
    for (int r = 0; r < 3; ++r)
        #pragma unroll
        for (int j = 0; j < 3; ++j) {
            float acc = 0.0f;
            #pragma unroll
            for (int s = 0; s < KS; ++s)
                acc += qkvp[(size_t)(r * KS + s) * DMODEL + 3 * h + j];
            M[r][j] = acc;
        }

    const float sc = 0.57735026918962576451f;  // 1/sqrt(3)
    float G[3][3];
    #pragma unroll
    for (int i = 0; i < 3; ++i)
        #pragma unroll
        for (int k = 0; k < 3; ++k)
            G[i][k] = (M[i][0] * M[k][0] + M[i][1] * M[k][1] + M[i][2] * M[k][2]) * sc;

    #pragma unroll
    for (int i = 0; i < 3; ++i)
        #pragma unroll
        for (int l = 0; l < 3; ++l)
            y[(size_t)h * 9 + i * 3 + l] =
                G[i][0] * M[0][l] + G[i][1] * M[1][l] + G[i][2] * M[2][l];
}

// ---------------------------------------------------------------------------
// Stage C: partial out:  poutp[(kz*3 + r)*D + c] = sum_{k in slice} y[r][k]*Wo[c][k]
// A tile streams 16 rows of Wo (NT); B columns 0..2 carry the three y rows
// (columns 3..15 zeroed). D tile column N=r holds out row r.
// ---------------------------------------------------------------------------
__global__ __launch_bounds__(128) void gemv_out_wmma(
    const float* __restrict__ y,     // (3, DMODEL) flat
    const float* __restrict__ Wo,
    float* __restrict__ poutp)       // (KS, 3, DMODEL) partials
{
    const int wave = threadIdx.x >> 5;
    const int lane = threadIdx.x & 31;
    const int half = lane >> 4;
    const int n    = lane & 15;      // tile column / Wo row-in-tile
    const int c0   = (blockIdx.x * 4 + wave) * 16;
    const int kz   = blockIdx.y;

    const float* wrow  = Wo + (size_t)(c0 + n) * DMODEL + half * 4;
    const int    yr    = (n < 3) ? n : 0;
    const float  keep  = (n < 3) ? 1.0f : 0.0f;
    const float* ybase = y + (size_t)yr * DMODEL + half * 4;
    const int kbeg = kz * KLEN;

    v8f c = {};
    #pragma unroll 8
    for (int k0 = kbeg; k0 < kbeg + KLEN; k0 += 8) {
        v4f a  = __builtin_nontemporal_load((const v4f*)(wrow + k0)); // streaming
        v4f yv = *(const v4f*)(ybase + k0);                           // L2-resident
        v2f a0; a0[0] = a[0];         a0[1] = a[1];
        v2f a1; a1[0] = a[2];         a1[1] = a[3];
        v2f b0; b0[0] = yv[0] * keep; b0[1] = yv[1] * keep;
        v2f b1; b1[0] = yv[2] * keep; b1[1] = yv[3] * keep;
        c = __builtin_amdgcn_wmma_f32_16x16x4_f32(false, a0, false, b0,
                                                  (short)0, c, false, false);
        c = __builtin_amdgcn_wmma_f32_16x16x4_f32(false, a1, false, b1,
                                                  (short)0, c, false, false);
    }
    // Lane r (r<3) of each half holds partial out[r][c0 + half*8 + v] in VGPR v.
    if (n < 3) {
        const int cbase = c0 + half * 8;
        float* p = poutp + (size_t)(kz * 3 + n) * DMODEL;
        #pragma unroll
        for (int v = 0; v < 8; ++v) p[cbase + v] = c[v];
    }
}

// ---------------------------------------------------------------------------
// Stage D: reduce K-split partials of the output projection + residual.
// out[r][c] = sum_s poutp[(s*3 + r)*D + c] + x[c]
// ---------------------------------------------------------------------------
__global__ __launch_bounds__(256) void reduce_out(
    const float* __restrict__ poutp,
    const float* __restrict__ xres,
    float* __restrict__ out)
{
    const int i = blockIdx.x * blockDim.x + threadIdx.x;   // 0 .. 3*D-1
    if (i >= 3 * DMODEL) return;
    const int r = i / DMODEL;
    const int c = i - r * DMODEL;
    float acc = xres[c];
    #pragma unroll
    for (int s = 0; s < KS; ++s)
        acc += poutp[(size_t)(s * 3 + r) * DMODEL + c];
    out[i] = acc;
}

extern "C" void kernel_launch(void* const* d_in, const int* in_sizes, int n_in,
                              void* d_out, int out_size, void* d_ws, size_t ws_size,
                              hipStream_t stream)
{
    const float* x  = (const float*)d_in[0];
    const float* Wq = (const float*)d_in[1];
    const float* Wk = (const float*)d_in[2];
    const float* Wv = (const float*)d_in[3];
    const float* Wo = (const float*)d_in[4];
    float* out = (float*)d_out;

    float* qkvp  = (float*)d_ws;                 // 3*KS*D floats  (288 KB)
    float* y     = qkvp + 3 * KS * DMODEL;       // 3*D floats     ( 72 KB)
    float* poutp = y + 3 * DMODEL;               // KS*3*D floats  (288 KB)

    dim3 gridA(DMODEL / 64, 3, KS);              // 4 waves/block, 16 rows/wave
    gemv_qkv_wmma<<<gridA, 128, 0, stream>>>(x, Wq, Wk, Wv, qkvp);
    head_attn_3x3<<<NHEAD / 256, 256, 0, stream>>>(qkvp, y);
    dim3 gridC(DMODEL / 64, KS);
    gemv_out_wmma<<<gridC, 128, 0, stream>>>(y, Wo, poutp);
    reduce_out<<<(3 * DMODEL + 255) / 256, 256, 0, stream>>>(poutp, x, out);
}